// GraphSAGE_18124761989810
// MI455X (gfx1250) — compile-verified
//
#include <hip/hip_runtime.h>
#include <math.h>

#define NNODES 100000
#define FDIM   128
#define CDIM   20

typedef float v2f __attribute__((ext_vector_type(2)));
typedef float v8f __attribute__((ext_vector_type(8)));

// ---------------------------------------------------------------- zero fill
__global__ void zero_f32(float* __restrict__ p, int n) {
  int i = blockIdx.x * blockDim.x + threadIdx.x;
  if (i < n) p[i] = 0.0f;
}

// ------------------------------------------------- edge gather + scatter-add
// One wave per edge. Lane l handles features {l, l+32, l+64, l+96}:
// fully coalesced 128B load of x[src] row and 128B of f32 atomics into
// agg[dst] row (L2-resident, 51MB << 192MB L2). Lane 0 counts degree.
__global__ void edge_scatter(const float* __restrict__ x,
                             const int* __restrict__ src,
                             const int* __restrict__ dst,
                             float* __restrict__ agg,
                             float* __restrict__ cnt,   // null => skip counting
                             int nedges) {
  const int wave = threadIdx.x >> 5;
  const int lane = threadIdx.x & 31;
  const int e = blockIdx.x * (blockDim.x >> 5) + wave;
  if (e >= nedges) return;
  const int s = src[e];
  const int d = dst[e];
  const float* __restrict__ xr = x + (size_t)s * FDIM;
  float* __restrict__ ar = agg + (size_t)d * FDIM;
#pragma unroll
  for (int i = 0; i < FDIM / 32; ++i) {
    atomicAdd(ar + lane + 32 * i, xr[lane + 32 * i]);
  }
  if (cnt != nullptr && lane == 0) atomicAdd(cnt + d, 1.0f);
}

// ------------------------------------------------------- fused SAGE GEMM
// out[row][n] = relu( bias[n]
//                   + sum_k (A[row][k] * rc(row)) * W1[n][k]      (mean path)
//                   + sum_k  X2[row][k]           * W2[n][k] )    (root path)
// Grid: one block per 16-row node tile; 8 waves = 8 column tiles of 16.
// Uses V_WMMA_F32_16X16X4_F32: A frag = float2 per lane (lanes 0-15: K=k,k+1;
// lanes 16-31: K=k+2,k+3; M = lane&15). B frag mirrors with N = lane&15, so
// W^T fragments are contiguous float2 loads from row-major W[n][k].
__global__ void sage_gemm_f32(const float* __restrict__ A,
                              const float* __restrict__ cnt,   // null => no mean scaling
                              const float* __restrict__ W1,    // [128][128] row-major
                              const float* __restrict__ bias,  // [128]
                              const float* __restrict__ X2,    // null => single-term GEMM
                              const float* __restrict__ W2,
                              float* __restrict__ out) {
  const int wave = threadIdx.x >> 5;
  const int lane = threadIdx.x & 31;
  const int lr = lane & 15;
  const int hi = lane >> 4;          // K-half select for A/B fragments
  const int row0 = blockIdx.x * 16;
  const int n0 = wave * 16;
  const int rowA = row0 + lr;

  float rc = 1.0f;
  if (cnt) rc = 1.0f / fmaxf(cnt[rowA], 1.0f);

  const float bval = bias[n0 + lr];  // per-column bias, same for all 8 acc regs
  v8f acc;
#pragma unroll
  for (int r = 0; r < 8; ++r) acc[r] = bval;

  const float* __restrict__ Ap = A + (size_t)rowA * FDIM + hi * 2;
  const float* __restrict__ Bp = W1 + (size_t)(n0 + lr) * FDIM + hi * 2;
#pragma unroll 4
  for (int k = 0; k < FDIM; k += 4) {
    v2f av = *(const v2f*)(Ap + k);
    av *= rc;                                    // fused mean division
    v2f bv = *(const v2f*)(Bp + k);
    acc = __builtin_amdgcn_wmma_f32_16x16x4_f32(
        false, av, false, bv, (short)0, acc, false, false);
  }

  if (X2) {
    const float* __restrict__ Ap2 = X2 + (size_t)rowA * FDIM + hi * 2;
    const float* __restrict__ Bp2 = W2 + (size_t)(n0 + lr) * FDIM + hi * 2;
#pragma unroll 4
    for (int k = 0; k < FDIM; k += 4) {
      v2f av = *(const v2f*)(Ap2 + k);
      v2f bv = *(const v2f*)(Bp2 + k);
      acc = __builtin_amdgcn_wmma_f32_16x16x4_f32(
          false, av, false, bv, (short)0, acc, false, false);
    }
  }

  // D layout: VGPR r, lanes 0-15 -> (M=row0+r, N=lr); lanes 16-31 -> M=row0+r+8
#pragma unroll
  for (int r = 0; r < 8; ++r) {
    const int m = row0 + r + hi * 8;
    out[(size_t)m * FDIM + n0 + lr] = fmaxf(acc[r], 0.0f);
  }
}

// ------------------------------------------------- final 128->20 + sigmoid
__global__ void mlp_out(const float* __restrict__ h,
                        const float* __restrict__ Wm2,  // [20][128]
                        const float* __restrict__ bm2,
                        float* __restrict__ out, int total) {
  int i = blockIdx.x * blockDim.x + threadIdx.x;
  if (i >= total) return;
  const int node = i / CDIM;
  const int c = i % CDIM;
  const float* __restrict__ hr = h + (size_t)node * FDIM;
  const float* __restrict__ wr = Wm2 + (size_t)c * FDIM;
  float s = bm2[c];
#pragma unroll 8
  for (int k = 0; k < FDIM; ++k) s += hr[k] * wr[k];
  out[i] = 1.0f / (1.0f + expf(-s));
}

// ---------------------------------------------------------------- launcher
extern "C" void kernel_launch(void* const* d_in, const int* in_sizes, int n_in,
                              void* d_out, int out_size, void* d_ws, size_t ws_size,
                              hipStream_t stream) {
  const float* x   = (const float*)d_in[0];
  const int*   ei  = (const int*)d_in[1];   // [2][E], row 0 = src, row 1 = dst
  const float* W1l = (const float*)d_in[2];
  const float* b1  = (const float*)d_in[3];
  const float* W1r = (const float*)d_in[4];
  const float* W2l = (const float*)d_in[5];
  const float* b2  = (const float*)d_in[6];
  const float* W2r = (const float*)d_in[7];
  const float* Wm1 = (const float*)d_in[8];
  const float* bm1 = (const float*)d_in[9];
  const float* Wm2 = (const float*)d_in[10];
  const float* bm2 = (const float*)d_in[11];
  float* out = (float*)d_out;

  const int E = in_sizes[1] / 2;
  const int n = NNODES;                      // == in_sizes[0] / FDIM
  const int nf = n * FDIM;

  // workspace layout: 3 N x 128 f32 buffers + degree counts
  float* buf0 = (float*)d_ws;                // agg1, later h2
  float* buf1 = buf0 + (size_t)n * FDIM;     // h1, later h3
  float* buf2 = buf1 + (size_t)n * FDIM;     // agg2
  float* cnt  = buf2 + (size_t)n * FDIM;     // degree (same for both layers)

  const dim3 b256(256);

  // ---- layer 1: aggregate + fused GEMM
  zero_f32<<<(nf + 255) / 256, b256, 0, stream>>>(buf0, nf);
  zero_f32<<<(n + 255) / 256, b256, 0, stream>>>(cnt, n);
  edge_scatter<<<(E + 7) / 8, b256, 0, stream>>>(x, ei, ei + E, buf0, cnt, E);
  sage_gemm_f32<<<n / 16, b256, 0, stream>>>(buf0, cnt, W1l, b1, x, W1r, buf1);

  // ---- layer 2: aggregate h1 + fused GEMM (degree unchanged -> reuse cnt)
  zero_f32<<<(nf + 255) / 256, b256, 0, stream>>>(buf2, nf);
  edge_scatter<<<(E + 7) / 8, b256, 0, stream>>>(buf1, ei, ei + E, buf2, nullptr, E);
  sage_gemm_f32<<<n / 16, b256, 0, stream>>>(buf2, cnt, W2l, b2, buf1, W2r, buf0);

  // ---- MLP head
  sage_gemm_f32<<<n / 16, b256, 0, stream>>>(buf0, nullptr, Wm1, bm1, nullptr, nullptr, buf1);
  const int total = n * CDIM;
  mlp_out<<<(total + 255) / 256, b256, 0, stream>>>(buf1, Wm2, bm2, out, total);
}